// GraphAttention_37443524887170
// MI455X (gfx1250) — compile-verified
//
#include <hip/hip_runtime.h>

// GAT layer for MI455X (gfx1250, wave32, WMMA + async LDS staging).
// N=4096 nodes, F_IN=256, F_OUT=64, H=8 heads, dense f32 adjacency.

typedef __attribute__((ext_vector_type(2))) float v2f;
typedef __attribute__((ext_vector_type(8))) float v8f;

#define NHEADS 8
#define NNODES 4096
#define FIN    256
#define FOUT   64
#define NEG_BIG -1e10f
#define LEAKY   0.2f

static __device__ __forceinline__ v8f wmma_f32_16x16x4(v2f a, v2f b, v8f c) {
  // D = A(16x4 f32) * B(4x16 f32) + C(16x16 f32)
  return __builtin_amdgcn_wmma_f32_16x16x4_f32(
      /*neg_a=*/false, a, /*neg_b=*/false, b,
      /*c_mod=*/(short)0, c, /*reuse_a=*/false, /*reuse_b=*/false);
}

// featsB staging layout: for head h, K-group jg=j/4, lane-half kh, col o, pair r:
//   featsB[((h*1024 + jg)*2 + kh)*128 + o*2 + r] = feats[h, 4*jg + 2*kh + r, o]
// so a per-lane b64 load yields the two B-operand VGPRs (K = v + 2*kh) directly.

// --------------------------------------------------------------------------
// Kernel 1: per-head projection feats = X @ W[h]  via f32 WMMA, staged store.
// grid = 256 row-tiles, block = 256 (8 waves; wave w handles head w).
__global__ __launch_bounds__(256) void k_project(const float* __restrict__ X,
                                                 const float* __restrict__ W,
                                                 float* __restrict__ featsB) {
  const int lane  = threadIdx.x & 31;
  const int h     = threadIdx.x >> 5;      // head per wave
  const int khalf = lane >> 4;
  const int lr    = lane & 15;
  const int i0    = blockIdx.x * 16;

  v8f c0 = {}, c1 = {}, c2 = {}, c3 = {};
  const float* xrow  = X + (size_t)(i0 + lr) * FIN + 2 * khalf;   // A operand rows
  const float* wbase = W + (size_t)h * (FIN * FOUT) + lr;

  for (int kb = 0; kb < FIN; kb += 4) {
    v2f a = *(const v2f*)(xrow + kb);                 // K = kb + 2*khalf + {0,1}
    const float* wk = wbase + (kb + 2 * khalf) * FOUT;
    v2f b0, b1, b2, b3;
    b0.x = wk[0];  b0.y = wk[FOUT + 0];
    b1.x = wk[16]; b1.y = wk[FOUT + 16];
    b2.x = wk[32]; b2.y = wk[FOUT + 32];
    b3.x = wk[48]; b3.y = wk[FOUT + 48];
    c0 = wmma_f32_16x16x4(a, b0, c0);
    c1 = wmma_f32_16x16x4(a, b1, c1);
    c2 = wmma_f32_16x16x4(a, b2, c2);
    c3 = wmma_f32_16x16x4(a, b3, c3);
  }

  // Scatter C (16x64 tile) into the B-staged layout.
#pragma unroll
  for (int r = 0; r < 8; ++r) {
    const int n    = i0 + r + 8 * khalf;              // node row of this C reg
    const int base = ((h * 1024 + (n >> 2)) * 2 + ((n >> 1) & 1)) * 128 + (n & 1);
    featsB[base + (lr +  0) * 2] = c0[r];
    featsB[base + (lr + 16) * 2] = c1[r];
    featsB[base + (lr + 32) * 2] = c2[r];
    featsB[base + (lr + 48) * 2] = c3[r];
  }
}

// --------------------------------------------------------------------------
// Kernel 2: a_s[h,n] = feats[h,n,:].ak_self[h],  a_n likewise. 32768 threads.
__global__ __launch_bounds__(256) void k_attncoef(const float* __restrict__ featsB,
                                                  const float* __restrict__ ak_self,
                                                  const float* __restrict__ ak_neigh,
                                                  float* __restrict__ a_s,
                                                  float* __restrict__ a_n) {
  const int t = blockIdx.x * 256 + threadIdx.x;       // 0 .. H*N-1
  const int h = t >> 12;
  const int n = t & (NNODES - 1);
  const int base = ((h * 1024 + (n >> 2)) * 2 + ((n >> 1) & 1)) * 128 + (n & 1);
  float as = 0.0f, an = 0.0f;
#pragma unroll 8
  for (int o = 0; o < FOUT; ++o) {
    const float f = featsB[base + o * 2];
    as = fmaf(f, ak_self[h * FOUT + o], as);
    an = fmaf(f, ak_neigh[h * FOUT + o], an);
  }
  a_s[t] = as;
  a_n[t] = an;
}

// --------------------------------------------------------------------------
// Kernel 3: fused masked-softmax attention + P@V via f32 WMMA.
// grid = 256 row-tiles, block = 256 (8 waves; wave w = head w).
// A tile (16 rows x 128 cols) double-buffered in LDS via async-to-LDS copies,
// shared by all 8 heads so the 64MB adjacency is read exactly once.
#define TILE_J   128
#define A_STRIDE 132   // dwords; 16B-aligned rows, bank-conflict-free

__global__ __launch_bounds__(256) void k_gat(const float* __restrict__ Adj,
                                             const float* __restrict__ featsB,
                                             const float* __restrict__ a_s,
                                             const float* __restrict__ a_n,
                                             const float* __restrict__ bias,
                                             float* __restrict__ out) {
  __shared__ float tileA[2][16 * A_STRIDE];

  const int lane  = threadIdx.x & 31;
  const int h     = threadIdx.x >> 5;
  const int khalf = lane >> 4;
  const int lr    = lane & 15;
  const int i0    = blockIdx.x * 16;

  const float as_row = a_s[h * NNODES + i0 + lr];     // A-operand row of this lane
  const float b0 = bias[h * FOUT + lr +  0];
  const float b1 = bias[h * FOUT + lr + 16];
  const float b2 = bias[h * FOUT + lr + 32];
  const float b3 = bias[h * FOUT + lr + 48];

  v8f acc0 = {}, acc1 = {}, acc2 = {}, acc3 = {};
  float lsum = 0.0f;

  // cooperative async A-tile loader: each of 256 threads moves 32B (2 x b128)
  const int arow = threadIdx.x >> 4;                  // 0..15 tile row
  const int acg  = threadIdx.x & 15;                  // 0..15 -> 8 dwords each
  const float* gbase = Adj + (size_t)(i0 + arow) * NNODES + acg * 8;
  const unsigned ldsOff0 =
      (unsigned)(uintptr_t)(void*)&tileA[0][arow * A_STRIDE + acg * 8];
  const unsigned ldsOff1 =
      (unsigned)(uintptr_t)(void*)&tileA[1][arow * A_STRIDE + acg * 8];

  // prologue: fill buffer 0 (IOFFSET applies to both LDS and global address)
  asm volatile(
      "global_load_async_to_lds_b128 %0, %1, off\n\t"
      "global_load_async_to_lds_b128 %0, %1, off offset:16"
      :: "v"(ldsOff0), "v"(gbase) : "memory");

  for (int jblock = 0; jblock < NNODES; jblock += TILE_J) {
    const int buf = (jblock >> 7) & 1;

    // my async copies (issued last iteration) done -> signal; barrier-exit
    // guarantees every wave's copies for `buf` are visible in LDS.
    asm volatile("s_wait_asynccnt 0x0" ::: "memory");
    __syncthreads();

    // prefetch next tile into the other buffer (overlaps with compute below)
    if (jblock + TILE_J < NNODES) {
      asm volatile(
          "global_load_async_to_lds_b128 %0, %1, off\n\t"
          "global_load_async_to_lds_b128 %0, %1, off offset:16"
          :: "v"(buf ? ldsOff0 : ldsOff1), "v"(gbase + jblock + TILE_J)
          : "memory");
    }

    const float* anp = a_n + h * NNODES + jblock + 2 * khalf;
    const float* tap = &tileA[buf][lr * A_STRIDE + 2 * khalf];

#pragma unroll 4
    for (int jb = 0; jb < TILE_J; jb += 4) {
      const int j = jblock + jb;
      const v2f an = *(const v2f*)(anp + jb);         // K = j + 2*khalf + {0,1}
      const v2f av = *(const v2f*)(tap + jb);         // adjacency, same K pair

      // leaky(a_s+a_n) + NEG_BIG*(1-A), then exp (exp(-1e10) -> 0 in fp32)
      const float s0 = as_row + an.x;
      const float s1 = as_row + an.y;
      const float e0 = __expf(fmaf(NEG_BIG, 1.0f - av.x, fmaxf(s0, LEAKY * s0)));
      const float e1 = __expf(fmaf(NEG_BIG, 1.0f - av.y, fmaxf(s1, LEAKY * s1)));
      lsum += e0 + e1;
      v2f p; p.x = e0; p.y = e1;

      const float* bb = featsB + (size_t)((h * 1024 + (j >> 2)) * 2 + khalf) * 128 + lr * 2;
      acc0 = wmma_f32_16x16x4(p, *(const v2f*)(bb +  0), acc0);
      acc1 = wmma_f32_16x16x4(p, *(const v2f*)(bb + 32), acc1);
      acc2 = wmma_f32_16x16x4(p, *(const v2f*)(bb + 64), acc2);
      acc3 = wmma_f32_16x16x4(p, *(const v2f*)(bb + 96), acc3);
    }
  }

  // fold the two K-halves of the row sums (rows live in lane r and r+16)
  lsum += __shfl_xor(lsum, 16, 32);

  // normalize, bias, relu, nontemporal store: out[n, h*64 + o]
#pragma unroll
  for (int r = 0; r < 8; ++r) {
    const float inv = 1.0f / __shfl(lsum, r + 8 * khalf, 32);
    const int n = i0 + r + 8 * khalf;
    float* orow = out + (size_t)n * (NHEADS * FOUT) + h * FOUT + lr;
    __builtin_nontemporal_store(fmaxf(fmaf(acc0[r], inv, b0), 0.0f), orow +  0);
    __builtin_nontemporal_store(fmaxf(fmaf(acc1[r], inv, b1), 0.0f), orow + 16);
    __builtin_nontemporal_store(fmaxf(fmaf(acc2[r], inv, b2), 0.0f), orow + 32);
    __builtin_nontemporal_store(fmaxf(fmaf(acc3[r], inv, b3), 0.0f), orow + 48);
  }
}

// --------------------------------------------------------------------------
extern "C" void kernel_launch(void* const* d_in, const int* in_sizes, int n_in,
                              void* d_out, int out_size, void* d_ws, size_t ws_size,
                              hipStream_t stream) {
  (void)in_sizes; (void)n_in; (void)out_size; (void)ws_size;
  const float* X        = (const float*)d_in[0];   // [4096, 256]
  const float* Adj      = (const float*)d_in[1];   // [4096, 4096]
  const float* W        = (const float*)d_in[2];   // [8, 256, 64]
  const float* bias     = (const float*)d_in[3];   // [8, 64]
  const float* ak_self  = (const float*)d_in[4];   // [8, 64]
  const float* ak_neigh = (const float*)d_in[5];   // [8, 64]
  float* out = (float*)d_out;                      // [4096, 512]

  float* featsB = (float*)d_ws;                                // 8 MB
  float* a_s    = featsB + (size_t)NHEADS * NNODES * FOUT;     // 128 KB
  float* a_n    = a_s + (size_t)NHEADS * NNODES;               // 128 KB

  k_project <<<NNODES / 16, 256, 0, stream>>>(X, W, featsB);
  k_attncoef<<<(NHEADS * NNODES) / 256, 256, 0, stream>>>(featsB, ak_self, ak_neigh, a_s, a_n);
  k_gat     <<<NNODES / 16, 256, 0, stream>>>(Adj, featsB, a_s, a_n, bias, out);
}